// RuGNN_54254026883316
// MI455X (gfx1250) — compile-verified
//
#include <hip/hip_runtime.h>
#include <math.h>

#define N_ENT  50000
#define N_EDGE 500000
#define H      128

typedef __attribute__((ext_vector_type(2))) float v2f;
typedef __attribute__((ext_vector_type(8))) float v8f;

// Float atomic max via signed/unsigned integer atomics (native global_atomic ops).
__device__ __forceinline__ void atomicMaxF(float* addr, float val) {
    if (val >= 0.0f)
        atomicMax((int*)addr, __float_as_int(val));
    else
        atomicMin((unsigned int*)addr, __float_as_uint(val));
}

// Native gfx1250 tanh (single TRANS op). Two v_nops cover the ISA's
// TRANS-result hazard window since the compiler can't see into the asm.
__device__ __forceinline__ float tanh_hw(float x) {
    float y;
    asm volatile("v_tanh_f32 %0, %1\n\tv_nop\n\tv_nop" : "=v"(y) : "v"(x));
    return y;
}

// ---------------------------------------------------------------- K0: init ws
__global__ void k_init(float* __restrict__ neigh, float* __restrict__ nmax,
                       float* __restrict__ nsum) {
    const long long total = 3LL * N_ENT * H;
    const long long stride = (long long)gridDim.x * blockDim.x;
    for (long long i = (long long)blockIdx.x * blockDim.x + threadIdx.x; i < total; i += stride)
        neigh[i] = 0.0f;
    for (int i = blockIdx.x * blockDim.x + threadIdx.x; i < 3 * N_ENT;
         i += gridDim.x * blockDim.x) {
        nmax[i] = -INFINITY;
        nsum[i] = 0.0f;
    }
}

// ------------------------------------------------ K1: per-edge logits + segmax
// One wave32 per edge; lane holds 4 features.  dotC = dotE + dotN.
__global__ void k_logits(const float* __restrict__ ent, const float* __restrict__ rel,
                         const int* __restrict__ src, const int* __restrict__ dst,
                         const int* __restrict__ rid,
                         float* __restrict__ nmax, float* __restrict__ logits) {
    const int e    = (blockIdx.x * blockDim.x + threadIdx.x) >> 5;  // exact: 62500*8 waves
    const int lane = threadIdx.x & 31;
    const int s = src[e], d = dst[e], r = rid[e];

    const float4 vv = *(const float4*)(ent + (long long)d * H + lane * 4);
    const float4 uu = *(const float4*)(ent + (long long)s * H + lane * 4);
    const float4 rr = *(const float4*)(rel + (long long)r * H + lane * 4);

    float dE = vv.x * rr.x + vv.y * rr.y + vv.z * rr.z + vv.w * rr.w;
    float dN = vv.x * uu.x + vv.y * uu.y + vv.z * uu.z + vv.w * uu.w;
#pragma unroll
    for (int off = 16; off > 0; off >>= 1) {
        dE += __shfl_xor(dE, off, 32);
        dN += __shfl_xor(dN, off, 32);
    }
    if (lane == 0) {
        const float dC = dE + dN;
        logits[0 * N_EDGE + e] = dE;
        logits[1 * N_EDGE + e] = dN;
        logits[2 * N_EDGE + e] = dC;
        atomicMaxF(nmax + 0 * N_ENT + d, dE);
        atomicMaxF(nmax + 1 * N_ENT + d, dN);
        atomicMaxF(nmax + 2 * N_ENT + d, dC);
    }
}

// --------------------------------------------------- K2: exp + segment sum
__global__ void k_expsum(const int* __restrict__ dst, const float* __restrict__ logits,
                         const float* __restrict__ nmax, float* __restrict__ nsum,
                         float* __restrict__ exv) {
    const int e = blockIdx.x * blockDim.x + threadIdx.x;
    if (e >= N_EDGE) return;
    const int d = dst[e];
#pragma unroll
    for (int l = 0; l < 3; ++l) {
        const float x = __expf(logits[l * N_EDGE + e] - nmax[l * N_ENT + d]);
        exv[l * N_EDGE + e] = x;
        atomicAdd(nsum + l * N_ENT + d, x);
    }
}

// ------------------------------------------- K3: weighted scatter into neigh
__global__ void k_scatter(const float* __restrict__ ent, const float* __restrict__ rel,
                          const int* __restrict__ src, const int* __restrict__ dst,
                          const int* __restrict__ rid,
                          const float* __restrict__ nsum, const float* __restrict__ exv,
                          float* __restrict__ neigh) {
    const int e    = (blockIdx.x * blockDim.x + threadIdx.x) >> 5;
    const int lane = threadIdx.x & 31;
    const int s = src[e], d = dst[e], r = rid[e];

    const float a0 = exv[0 * N_EDGE + e] / nsum[0 * N_ENT + d];
    const float a1 = exv[1 * N_EDGE + e] / nsum[1 * N_ENT + d];
    const float a2 = exv[2 * N_EDGE + e] / nsum[2 * N_ENT + d];

    const float4 uu = *(const float4*)(ent + (long long)s * H + lane * 4);
    const float4 rr = *(const float4*)(rel + (long long)r * H + lane * 4);

    float* p0 = neigh + ((long long)(0 * N_ENT + d)) * H + lane * 4;
    float* p1 = neigh + ((long long)(1 * N_ENT + d)) * H + lane * 4;
    float* p2 = neigh + ((long long)(2 * N_ENT + d)) * H + lane * 4;

    atomicAdd(p0 + 0, rr.x * a0); atomicAdd(p0 + 1, rr.y * a0);
    atomicAdd(p0 + 2, rr.z * a0); atomicAdd(p0 + 3, rr.w * a0);

    atomicAdd(p1 + 0, uu.x * a1); atomicAdd(p1 + 1, uu.y * a1);
    atomicAdd(p1 + 2, uu.z * a1); atomicAdd(p1 + 3, uu.w * a1);

    atomicAdd(p2 + 0, (uu.x + rr.x) * a2); atomicAdd(p2 + 1, (uu.y + rr.y) * a2);
    atomicAdd(p2 + 2, (uu.z + rr.z) * a2); atomicAdd(p2 + 3, (uu.w + rr.w) * a2);
}

// ---------------------------- K4: 3x (neigh @ W) via WMMA f32 + tanh + residual
#define SA_STRIDE (H + 1)  // 129: avoids LDS bank conflicts on column reads

__global__ __launch_bounds__(256) void k_gemm(const float* __restrict__ neigh,
                                              const float* __restrict__ ent,
                                              const float* __restrict__ wE,
                                              const float* __restrict__ wN,
                                              const float* __restrict__ wC,
                                              float* __restrict__ out) {
    __shared__ float sA[3 * 16 * SA_STRIDE];

    const int m0  = blockIdx.x * 16;
    const int tid = threadIdx.x;

    // Stage the three 16x128 A tiles (one per layer) into LDS.
    for (int idx = tid; idx < 3 * 16 * H; idx += 256) {
        const int l   = idx >> 11;          // / 2048
        const int rem = idx & 2047;
        const int row = rem >> 7;           // / 128
        const int col = rem & 127;
        sA[(l * 16 + row) * SA_STRIDE + col] =
            neigh[((long long)(l * N_ENT + m0 + row)) * H + col];
    }
    __syncthreads();

    const int lane = tid & 31;
    const int w    = tid >> 5;      // 8 waves -> 8 column tiles
    const int n0   = w * 16;
    const int half = lane >> 4;     // 0: K={0,1}, 1: K={2,3} within each x4 step
    const int mr   = lane & 15;

    const float* Ws[3] = {wE, wN, wC};
    float tot[8] = {0.f, 0.f, 0.f, 0.f, 0.f, 0.f, 0.f, 0.f};

#pragma unroll
    for (int l = 0; l < 3; ++l) {
        const float* W = Ws[l];
        const float* A = sA + l * 16 * SA_STRIDE;
        v8f c = {0.f, 0.f, 0.f, 0.f, 0.f, 0.f, 0.f, 0.f};
#pragma unroll 4
        for (int k0 = 0; k0 < H; k0 += 4) {
            const int k = k0 + 2 * half;
            v2f a;  // A 16x4: lane holds row M=mr, K = {k, k+1}
            a.x = A[mr * SA_STRIDE + k];
            a.y = A[mr * SA_STRIDE + k + 1];
            v2f b;  // B 4x16: lane holds col N=mr, K = {k, k+1}
            b.x = W[k * H + n0 + mr];
            b.y = W[(k + 1) * H + n0 + mr];
            c = __builtin_amdgcn_wmma_f32_16x16x4_f32(
                    false, a, false, b, (short)0, c, false, false);
        }
#pragma unroll
        for (int rr = 0; rr < 8; ++rr) tot[rr] += tanh_hw(c[rr]);
    }

    // D layout: VGPR rr -> row (rr + 8*half), col = n0 + mr.
#pragma unroll
    for (int rr = 0; rr < 8; ++rr) {
        const long long row = m0 + rr + 8 * half;
        const long long idx = row * H + n0 + mr;
        out[idx] = ent[idx] + tot[rr];
    }
}

// ---------------------------------------------------------------- launcher
extern "C" void kernel_launch(void* const* d_in, const int* in_sizes, int n_in,
                              void* d_out, int out_size, void* d_ws, size_t ws_size,
                              hipStream_t stream) {
    (void)in_sizes; (void)n_in; (void)out_size; (void)ws_size;

    const float* ent = (const float*)d_in[0];
    const float* rel = (const float*)d_in[1];
    const float* wE  = (const float*)d_in[2];
    const float* wN  = (const float*)d_in[3];
    const float* wC  = (const float*)d_in[4];
    const int*   src = (const int*)d_in[5];
    const int*   dst = (const int*)d_in[6];
    const int*   rid = (const int*)d_in[7];
    float*       out = (float*)d_out;

    float* ws     = (float*)d_ws;
    float* neigh  = ws;                         // 3*N*H
    float* nmax   = neigh + 3LL * N_ENT * H;    // 3*N
    float* nsum   = nmax + 3 * N_ENT;           // 3*N
    float* logits = nsum + 3 * N_ENT;           // 3*E
    float* exv    = logits + 3 * N_EDGE;        // 3*E

    k_init<<<2048, 256, 0, stream>>>(neigh, nmax, nsum);
    k_logits<<<N_EDGE / 8, 256, 0, stream>>>(ent, rel, src, dst, rid, nmax, logits);
    k_expsum<<<(N_EDGE + 255) / 256, 256, 0, stream>>>(dst, logits, nmax, nsum, exv);
    k_scatter<<<N_EDGE / 8, 256, 0, stream>>>(ent, rel, src, dst, rid, nsum, exv, neigh);
    k_gemm<<<N_ENT / 16, 256, 0, stream>>>(neigh, ent, wE, wN, wC, out);
}